// MoNetLayer_11751030521976
// MI455X (gfx1250) — compile-verified
//
#include <hip/hip_runtime.h>
#include <hip/hip_bf16.h>

// MoNet layer for MI455X (gfx1250).
// Bound by scattered mu/sig gathers + f32 atomics, not FLOPs. Strategy:
//  - transpose mu/sig (C,N,3) -> node-major (N,C,6) with 1/sigma folded in:
//    76.8 MB, fits MI455X's 192 MB L2 -> edge-phase gathers become L2 hits.
//  - per-edge 3x3 affine (u = tanh(cd@W+b)) done 16 edges/wave with
//    V_WMMA_F32_16X16X4_F32 (exact fp32).
//  - one wave per edge; lane c handles Gaussian kernel c; 32 contiguous
//    global_atomic_add_f32 per edge for the segment-sum.

typedef __attribute__((ext_vector_type(2))) float v2f;
typedef __attribute__((ext_vector_type(8))) float v8f;

#define DIMC 3

// ---------------------------------------------------------------- init + hsum
// One wave per node: copy coords into out[:, :3], zero out[:, 3:3+C]
// (required every call: harness poisons d_out and we accumulate atomically),
// and wave-reduce hsum[n] = sum_f h[n, f].
__global__ void init_hsum_kernel(const float* __restrict__ feat,
                                 float* __restrict__ out,
                                 float* __restrict__ hsum,
                                 int N, int FL, int C) {
  int lane = threadIdx.x & 31;
  int node = blockIdx.x * (blockDim.x >> 5) + (threadIdx.x >> 5);
  if (node >= N) return;
  const float* f = feat + (size_t)node * FL;
  float* orow = out + (size_t)node * (DIMC + C);
  if (lane < DIMC) orow[lane] = f[lane];
  for (int c = lane; c < C; c += 32) orow[DIMC + c] = 0.0f;
  if (hsum) {
    float s = 0.0f;
    for (int k = lane; k < FL - DIMC; k += 32) s += f[DIMC + k];
    for (int off = 16; off; off >>= 1) s += __shfl_xor(s, off);
    if (lane == 0) hsum[node] = s;
  }
}

// ------------------------------------------------- mu/sig transpose to (N,C,6)
// Block handles 8 nodes x C kernels. Reads are coalesced 24-float chunks of
// the (C,N,3) layout, staged through LDS, written as one contiguous
// 8*C*6-float block in node-major order. sigma stored as reciprocal.
__global__ void transpose_kernel(const float* __restrict__ mu,
                                 const float* __restrict__ sig,
                                 float* __restrict__ mst,
                                 int N, int C) {
  extern __shared__ float tile[];
  const int NB = 8;
  int n0 = blockIdx.x * NB;
  int total = NB * C * 6;
  for (int idx = threadIdx.x; idx < total; idx += blockDim.x) {
    int chunk = idx / (NB * 3);   // 0 .. 2C-1  (c, mu-or-sig)
    int off   = idx % (NB * 3);   // 0 .. 23    contiguous in global memory
    int c     = chunk >> 1;
    int isSig = chunk & 1;
    int n = off / 3, r = off % 3;
    int node = n0 + n;
    float v = 1.0f;
    if (node < N) {
      size_t ix = ((size_t)c * N + node) * 3 + r;
      v = isSig ? (1.0f / sig[ix]) : mu[ix];
    }
    tile[(n * C + c) * 6 + isSig * 3 + r] = v;
  }
  __syncthreads();
  float* dst = mst + (size_t)n0 * C * 6;
  for (int j = threadIdx.x; j < total; j += blockDim.x) {
    int node = n0 + j / (C * 6);
    if (node < N) dst[j] = tile[j];
  }
}

// -------------------------------------------------------- u = tanh(cd@W + b)
// 16 edges per wave via V_WMMA_F32_16X16X4_F32.
// A (16x4, M=edge, K=dim):   lane m<16 holds {A[m][0],A[m][1]},
//                            lane 16+m holds {A[m][2],A[m][3]=0}.
// B (4x16, K=dim, N=outdim): lane n<16 holds {B[0][n],B[1][n]},
//                            lane 16+n holds {B[2][n],B[3][n]=0}.
// D (16x16): VGPR j, lane n<16 -> D[j][n]; lane 16+n -> D[8+j][n].
// Both wave halves load the same 16 edges, so no cross-lane shuffles needed.
// W/b loads are unconditional with clamped index (no exec-mask branches);
// zeros are materialized with selects instead of guarded loads.
__global__ void u_wmma_kernel(const float* __restrict__ feat,
                              const int* __restrict__ ei,
                              const float* __restrict__ W,
                              const float* __restrict__ b,
                              float* __restrict__ u_ws,
                              int E, int FL) {
  int lane = threadIdx.x & 31;
  int wib  = threadIdx.x >> 5;
  int half = lane >> 4;
  int m    = lane & 15;
  long e0 = ((long)blockIdx.x * 8 + wib) * 16;
  if (e0 >= E) return;                 // wave-uniform: EXEC stays all-ones
  long e = e0 + m;
  if (e >= E) e = E - 1;               // clamp (select, no divergence)
  int2 sd = ((const int2*)ei)[e];      // one global_load_b64
  const float* fs = feat + (size_t)sd.x * FL;
  const float* fd = feat + (size_t)sd.y * FL;
  float cd0 = fd[0] - fs[0];
  float cd1 = fd[1] - fs[1];
  float cd2 = fd[2] - fs[2];

  v2f a, bb;
  a[0] = half ? cd2 : cd0;
  a[1] = half ? 0.0f : cd1;

  bool act = (m < DIMC);
  int n3 = act ? m : 0;                // clamped column index
  float w0 = W[n3];                    // unconditional broadcast loads
  float w1 = W[DIMC + n3];
  float w2 = W[2 * DIMC + n3];
  bb[0] = act ? (half ? w2 : w0) : 0.0f;
  bb[1] = (act && !half) ? w1 : 0.0f;

  v8f c = {};
  v8f d = __builtin_amdgcn_wmma_f32_16x16x4_f32(
      false, a, false, bb, (short)0, c, false, false);

  float bias = act ? b[n3] : 0.0f;
#pragma unroll
  for (int j = 0; j < 8; ++j) {
    int row = half ? (8 + j) : j;
    long eo = e0 + row;
    if (act && eo < E) u_ws[eo * 4 + m] = tanhf(d[j] + bias);
  }
}

// ------------------------------------------------------------------ edge pass
// One wave per edge; lane c = Gaussian kernel index. TRANS: mu/rsig from the
// node-major L2-resident repack (three b64 loads = coalesced 768 B per wave).
// C32 fast path removes the per-lane loop. Fallback paths read the raw
// (C,N,3) layout / recompute u / recompute hsum inline.
template <bool TRANS, bool UWS, bool HS, bool C32>
__global__ void edge_kernel(const int* __restrict__ ei,
                            const float* __restrict__ feat,
                            const float* __restrict__ W,
                            const float* __restrict__ b,
                            const float* __restrict__ u_ws,
                            const float* __restrict__ hsum,
                            const float* __restrict__ mu,
                            const float* __restrict__ sig,
                            const float* __restrict__ mst,
                            float* __restrict__ out,
                            int N, int E, int C, int FL) {
  int wid = blockIdx.x * (blockDim.x >> 5) + (threadIdx.x >> 5);
  if (wid >= E) return;
  int lane = threadIdx.x & 31;
  long e = wid;
  int2 sd = ((const int2*)ei)[e];      // one global_load_b64
  int src = sd.x, dst = sd.y;

  float u0, u1, u2;
  if (UWS) {
    float4 uv = ((const float4*)u_ws)[e];  // one global_load_b128
    u0 = uv.x; u1 = uv.y; u2 = uv.z;
  } else {
    const float* fs = feat + (size_t)src * FL;
    const float* fd = feat + (size_t)dst * FL;
    float c0 = fd[0] - fs[0], c1 = fd[1] - fs[1], c2 = fd[2] - fs[2];
    u0 = tanhf(c0 * W[0] + c1 * W[3] + c2 * W[6] + b[0]);
    u1 = tanhf(c0 * W[1] + c1 * W[4] + c2 * W[7] + b[1]);
    u2 = tanhf(c0 * W[2] + c1 * W[5] + c2 * W[8] + b[2]);
  }

  float hs;
  if (HS) {
    hs = hsum[dst];
  } else {
    float s = 0.0f;
    const float* fh = feat + (size_t)dst * FL + DIMC;
    for (int k = lane; k < FL - DIMC; k += 32) s += fh[k];
    for (int off = 16; off; off >>= 1) s += __shfl_xor(s, off);
    hs = s;
  }

  float* orow = out + (size_t)src * (DIMC + C) + DIMC;
  int cEnd = C32 ? (lane + 1) : C;     // C32: exactly one iteration
  for (int c = lane; c < cEnd; c += 32) {
    float m0, m1, m2, r0, r1, r2;
    if (TRANS) {
      const float2* p = (const float2*)(mst + ((size_t)src * C + c) * 6);
      float2 ab = p[0], cd2 = p[1], ef = p[2];   // three global_load_b64
      m0 = ab.x; m1 = ab.y; m2 = cd2.x;
      r0 = cd2.y; r1 = ef.x; r2 = ef.y;
    } else {
      size_t ix = ((size_t)c * N + src) * 3;
      m0 = mu[ix]; m1 = mu[ix + 1]; m2 = mu[ix + 2];
      r0 = 1.0f / sig[ix]; r1 = 1.0f / sig[ix + 1]; r2 = 1.0f / sig[ix + 2];
    }
    float d0 = u0 - m0, d1 = u1 - m1, d2 = u2 - m2;
    float wgt = expf(-0.5f * (d0 * d0 * r0 + d1 * d1 * r1 + d2 * d2 * r2));
    atomicAdd(orow + c, wgt * hs);   // 32 contiguous f32 atomics, L2-resident
  }
}

// ------------------------------------------------------------------- launcher
extern "C" void kernel_launch(void* const* d_in, const int* in_sizes, int n_in,
                              void* d_out, int out_size, void* d_ws, size_t ws_size,
                              hipStream_t stream) {
  const float* features = (const float*)d_in[0];
  const int*   ei       = (const int*)d_in[1];
  const float* W        = (const float*)d_in[2];
  const float* b        = (const float*)d_in[3];
  const float* mu       = (const float*)d_in[4];
  const float* sig      = (const float*)d_in[5];
  float* out = (float*)d_out;

  // Recover shapes: |mu| = C*N*3, out = N*(3+C), |features| = N*FL, |edges|=2E
  long NC = (long)in_sizes[4] / 3;
  int  N  = (int)(((long)out_size - NC) / DIMC);
  int  C  = (int)(NC / N);
  int  FL = in_sizes[0] / N;
  int  E  = in_sizes[1] / 2;

  size_t uOff  = ((size_t)N + 3) & ~(size_t)3;    // keep u_ws 16B-aligned
  size_t needH = (size_t)N * sizeof(float);                       // hsum
  size_t needU = (uOff + (size_t)E * 4) * sizeof(float);          // + u (E x 4)
  size_t needT = needU + (size_t)N * C * 6 * sizeof(float);       // + mu/rsig repack
  bool haveH = ws_size >= needH;
  bool haveU = ws_size >= needU;
  bool haveT = ws_size >= needT;

  float* ws   = (float*)d_ws;
  float* hsum = haveH ? ws : nullptr;
  float* u_ws = ws + uOff;
  float* mst  = ws + uOff + (size_t)E * 4;

  int nodeBlocks = (N + 7) / 8;
  init_hsum_kernel<<<nodeBlocks, 256, 0, stream>>>(features, out, hsum, N, FL, C);

  if (haveT) {
    int smem = 8 * C * 6 * (int)sizeof(float);
    transpose_kernel<<<nodeBlocks, 256, smem, stream>>>(mu, sig, mst, N, C);
  }
  if (haveU) {
    int ublocks = (E + 127) / 128;  // 8 waves/block, 16 edges/wave
    u_wmma_kernel<<<ublocks, 256, 0, stream>>>(features, ei, W, b, u_ws, E, FL);
  }

  int eblocks = (E + 7) / 8;        // 8 waves/block, 1 edge/wave
  if (haveT && C == 32)
    edge_kernel<true, true, true, true><<<eblocks, 256, 0, stream>>>(
        ei, features, W, b, u_ws, hsum, mu, sig, mst, out, N, E, C, FL);
  else if (haveT)
    edge_kernel<true, true, true, false><<<eblocks, 256, 0, stream>>>(
        ei, features, W, b, u_ws, hsum, mu, sig, mst, out, N, E, C, FL);
  else if (haveU)
    edge_kernel<false, true, true, false><<<eblocks, 256, 0, stream>>>(
        ei, features, W, b, u_ws, hsum, mu, sig, mst, out, N, E, C, FL);
  else if (haveH)
    edge_kernel<false, false, true, false><<<eblocks, 256, 0, stream>>>(
        ei, features, W, b, u_ws, hsum, mu, sig, mst, out, N, E, C, FL);
  else
    edge_kernel<false, false, false, false><<<eblocks, 256, 0, stream>>>(
        ei, features, W, b, u_ws, hsum, mu, sig, mst, out, N, E, C, FL);
}